// GraphQ_1984274891291
// MI455X (gfx1250) — compile-verified
//
#include <hip/hip_runtime.h>
#include <hip/hip_bf16.h>

// GCN Q-network forward for MI455X (gfx1250, wave32).
// GEMMs use V_WMMA_F32_16X16X4_F32 (exact fp32 semantics, matrix pipe), with
// each wave computing a 16x64 output strip (4 independent accumulators) to
// hide the WMMA->WMMA RAW latency and amortize A-fragment loads 4x.
// Edge scatter relies on 192MB L2 holding hw/acc (51.2MB each) so the
// random-index traffic never hits HBM; f32 atomics via global_atomic_add_f32.
// The reference's `e = edge_attr@We+be` is dead code and is skipped.

typedef __attribute__((ext_vector_type(2))) float v2f;
typedef __attribute__((ext_vector_type(8))) float v8f;

#define HIDDIM 256

// ---------------------------------------------------------------------------
// fp32 WMMA GEMM:  C[M,256] = op(A[M,K] @ B[K,256] + bias (+ extra))
// blockDim = 128 (4 waves). Block owns one 16-row tile x all 256 cols.
// Wave w owns a 16x64 strip: 4 col tiles at cols (w*4 + j)*16, j = 0..3.
// Per k-step: one A-fragment (b64 load) feeds 4 WMMAs on 4 accumulators.
// Requires: M % 16 == 0, K % 4 == 0, Ncols == 256. EXEC stays all-ones.
// extra_vec/extra_w (optional): C[r,c] += extra_vec[r] * extra_w[c]
//   (folds the concat([Z, action]) 257th K-column into the epilogue).
// ---------------------------------------------------------------------------
__global__ void gemm_wmma_f32(const float* __restrict__ A,
                              const float* __restrict__ B,
                              const float* __restrict__ bias,
                              float* __restrict__ C,
                              int M, int K, int Ncols, int relu,
                              const float* __restrict__ extra_vec,
                              const float* __restrict__ extra_w) {
  const int wave = threadIdx.x >> 5;
  const int lane = threadIdx.x & 31;
  const int half = lane >> 4;      // 0: lanes 0-15 (K=0,1), 1: lanes 16-31 (K=2,3)
  const int m    = lane & 15;      // M index for A frag, N index for B frag
  const int row0 = blockIdx.x * 16;
  const int col0 = wave * 64;      // 16x64 strip per wave

  const float* Arow = A + (size_t)(row0 + m) * K;            // A[16x4]: V0=K+2h, V1=K+2h+1
  const float* Bp   = B + (size_t)(2 * half) * Ncols + col0 + m; // B[4x16]: same K split

  v8f acc0 = {}, acc1 = {}, acc2 = {}, acc3 = {};
  for (int k = 0; k < K; k += 4) {
    v2f a;
    a.x = Arow[k + 2 * half];
    a.y = Arow[k + 2 * half + 1];
    const float* b0 = Bp + (size_t)k * Ncols;
    const float* b1 = Bp + (size_t)(k + 1) * Ncols;
    v2f bA, bB, bC, bD;
    bA.x = b0[0];  bA.y = b1[0];
    bB.x = b0[16]; bB.y = b1[16];
    bC.x = b0[32]; bC.y = b1[32];
    bD.x = b0[48]; bD.y = b1[48];
    // 8 args: (neg_a, A, neg_b, B, c_mod, C, reuse_a, reuse_b)
    acc0 = __builtin_amdgcn_wmma_f32_16x16x4_f32(false, a, false, bA, (short)0, acc0, false, false);
    acc1 = __builtin_amdgcn_wmma_f32_16x16x4_f32(false, a, false, bB, (short)0, acc1, false, false);
    acc2 = __builtin_amdgcn_wmma_f32_16x16x4_f32(false, a, false, bC, (short)0, acc2, false, false);
    acc3 = __builtin_amdgcn_wmma_f32_16x16x4_f32(false, a, false, bD, (short)0, acc3, false, false);
  }

  float bv[4] = {0.f, 0.f, 0.f, 0.f};
  float ew[4] = {0.f, 0.f, 0.f, 0.f};
#pragma unroll
  for (int j = 0; j < 4; ++j) {
    if (bias)      bv[j] = bias[col0 + j * 16 + m];
    if (extra_vec) ew[j] = extra_w[col0 + j * 16 + m];
  }
#pragma unroll
  for (int r = 0; r < 8; ++r) {
    const int rr = row0 + r + 8 * half;   // C/D layout: VGPR r -> M = r + 8*half
    const float evr = extra_vec ? extra_vec[rr] : 0.0f;
    float v0 = acc0[r] + bv[0] + evr * ew[0];
    float v1 = acc1[r] + bv[1] + evr * ew[1];
    float v2 = acc2[r] + bv[2] + evr * ew[2];
    float v3 = acc3[r] + bv[3] + evr * ew[3];
    if (relu) {
      v0 = fmaxf(v0, 0.0f); v1 = fmaxf(v1, 0.0f);
      v2 = fmaxf(v2, 0.0f); v3 = fmaxf(v3, 0.0f);
    }
    float* Crow = C + (size_t)rr * Ncols + col0 + m;
    Crow[0]  = v0;
    Crow[16] = v1;
    Crow[32] = v2;
    Crow[48] = v3;
  }
}

// ---------------------------------------------------------------------------
// Degree / normalization
// ---------------------------------------------------------------------------
__global__ void deg_kernel(const int* __restrict__ dst, float* __restrict__ deg, int E) {
  int i = blockIdx.x * blockDim.x + threadIdx.x;
  if (i < E) atomicAdd(&deg[dst[i]], 1.0f);
}

__global__ void rsqrt_kernel(float* __restrict__ dis, int N) {
  int i = blockIdx.x * blockDim.x + threadIdx.x;
  if (i < N) dis[i] = rsqrtf(dis[i] + 1.0f);   // +1 self loop
}

// ---------------------------------------------------------------------------
// Edge scatter: acc[dst] += dis[src]*dis[dst] * hw[src]   (one wave per edge)
// hw/acc are L2-resident (51.2MB each << 192MB L2).
// ---------------------------------------------------------------------------
__global__ void scatter_kernel(const float* __restrict__ hw,
                               const int* __restrict__ src,
                               const int* __restrict__ dst,
                               const float* __restrict__ dis,
                               float* __restrict__ acc, int E) {
  const int wave = (blockIdx.x * blockDim.x + threadIdx.x) >> 5;
  const int lane = threadIdx.x & 31;
  if (wave >= E) return;
  const int s = src[wave];
  const int d = dst[wave];
  const float norm = dis[s] * dis[d];
  const float4* hs = (const float4*)(hw + (size_t)s * HIDDIM);
  float* ad = acc + (size_t)d * HIDDIM;
#pragma unroll
  for (int c = 0; c < 2; ++c) {
    const int q = lane + c * 32;       // 64 float4 chunks per row
    float4 v = hs[q];
    const int f = q * 4;
    atomicAdd(ad + f + 0, norm * v.x);
    atomicAdd(ad + f + 1, norm * v.y);
    atomicAdd(ad + f + 2, norm * v.z);
    atomicAdd(ad + f + 3, norm * v.w);
  }
}

// H = relu(acc + dis^2 * hw + bias)   (self-loop message + bias + activation)
__global__ void combine_kernel(const float* __restrict__ acc,
                               const float* __restrict__ hw,
                               const float* __restrict__ dis,
                               const float* __restrict__ bias,
                               float* __restrict__ H) {
  const int row = blockIdx.x;
  const int f   = threadIdx.x;
  const size_t i = (size_t)row * HIDDIM + f;
  const float d2 = dis[row] * dis[row];
  H[i] = fmaxf(acc[i] + d2 * hw[i] + bias[f], 0.0f);
}

// Z[r,:] = H[map[r],:]
__global__ void gather_kernel(const float* __restrict__ H,
                              const int* __restrict__ map,
                              float* __restrict__ Z) {
  const int r = blockIdx.x;
  const int f = threadIdx.x;
  Z[(size_t)r * HIDDIM + f] = H[(size_t)map[r] * HIDDIM + f];
}

// q = mean(Z2, axis=0) @ Wh + bh  -> out[0]   (single block, 256 threads)
__global__ void reduce_q_kernel(const float* __restrict__ Z2,
                                const float* __restrict__ Wh,
                                const float* __restrict__ bh,
                                float* __restrict__ out, int NS) {
  const int t = threadIdx.x;
  float s = 0.0f;
  for (int r = 0; r < NS; ++r) s += Z2[(size_t)r * HIDDIM + t];
  s = (s / (float)NS) * Wh[t];
  __shared__ float sm[HIDDIM];
  sm[t] = s;
  __syncthreads();
  for (int step = HIDDIM / 2; step > 0; step >>= 1) {
    if (t < step) sm[t] += sm[t + step];
    __syncthreads();
  }
  if (t == 0) out[0] = sm[0] + bh[0];
}

// ---------------------------------------------------------------------------
extern "C" void kernel_launch(void* const* d_in, const int* in_sizes, int n_in,
                              void* d_out, int out_size, void* d_ws, size_t ws_size,
                              hipStream_t stream) {
  (void)n_in; (void)out_size; (void)ws_size;

  const float* x      = (const float*)d_in[0];
  const float* action = (const float*)d_in[2];
  const float* Wn  = (const float*)d_in[3];  const float* bn  = (const float*)d_in[4];
  // d_in[5]/d_in[6] (We/be): dead code in the reference forward — skipped.
  const float* Wc1 = (const float*)d_in[7];  const float* bc1 = (const float*)d_in[8];
  const float* Wc2 = (const float*)d_in[9];  const float* bc2 = (const float*)d_in[10];
  const float* Wm1 = (const float*)d_in[11]; const float* bm1 = (const float*)d_in[12];
  const float* Wm2 = (const float*)d_in[13]; const float* bm2 = (const float*)d_in[14];
  const float* Wh  = (const float*)d_in[15]; const float* bh  = (const float*)d_in[16];
  const int* edge_index = (const int*)d_in[17];
  const int* sgen       = (const int*)d_in[18];

  const int N  = in_sizes[0] / 32;     // 50000 (multiple of 16)
  const int E  = in_sizes[17] / 2;     // 800000
  const int NS = in_sizes[18];         // 4096
  const int* src = edge_index;
  const int* dst = edge_index + E;

  // Workspace layout (zero-init handled per-launch below where required)
  char* ws = (char*)d_ws;
  size_t off = 0;
  auto wsalloc = [&](size_t bytes) -> void* {
    void* p = ws + off;
    off += (bytes + 255) & ~(size_t)255;
    return p;
  };
  float* H   = (float*)wsalloc((size_t)N  * HIDDIM * sizeof(float));
  float* HW  = (float*)wsalloc((size_t)N  * HIDDIM * sizeof(float));
  float* ACC = (float*)wsalloc((size_t)N  * HIDDIM * sizeof(float));
  float* DIS = (float*)wsalloc((size_t)N  * sizeof(float));
  float* Z   = (float*)wsalloc((size_t)NS * HIDDIM * sizeof(float)); // Z, later reused as Z2
  float* Z1  = (float*)wsalloc((size_t)NS * HIDDIM * sizeof(float));

  const dim3 gemmBlk(128);             // 4 waves, one 16x256 row tile per block
  const dim3 gemmGridN(N / 16);        // conv-size GEMMs
  const dim3 gemmGridS(NS / 16);       // MLP GEMMs

  // --- degrees / symmetric normalization (shared by both conv layers) ---
  hipMemsetAsync(DIS, 0, (size_t)N * sizeof(float), stream);
  deg_kernel<<<(E + 255) / 256, 256, 0, stream>>>(dst, DIS, E);
  rsqrt_kernel<<<(N + 255) / 256, 256, 0, stream>>>(DIS, N);

  // --- h = x @ Wn + bn ---
  gemm_wmma_f32<<<gemmGridN, gemmBlk, 0, stream>>>(x, Wn, bn, H, N, 32, HIDDIM, 0,
                                                   nullptr, nullptr);

  // --- conv layer 1: h = relu(GCNConv(h; Wc1, bc1)) ---
  gemm_wmma_f32<<<gemmGridN, gemmBlk, 0, stream>>>(H, Wc1, nullptr, HW, N, HIDDIM,
                                                   HIDDIM, 0, nullptr, nullptr);
  hipMemsetAsync(ACC, 0, (size_t)N * HIDDIM * sizeof(float), stream);
  scatter_kernel<<<(E + 7) / 8, 256, 0, stream>>>(HW, src, dst, DIS, ACC, E);
  combine_kernel<<<N, HIDDIM, 0, stream>>>(ACC, HW, DIS, bc1, H);

  // --- conv layer 2: h = relu(GCNConv(h; Wc2, bc2)) ---
  gemm_wmma_f32<<<gemmGridN, gemmBlk, 0, stream>>>(H, Wc2, nullptr, HW, N, HIDDIM,
                                                   HIDDIM, 0, nullptr, nullptr);
  hipMemsetAsync(ACC, 0, (size_t)N * HIDDIM * sizeof(float), stream);
  scatter_kernel<<<(E + 7) / 8, 256, 0, stream>>>(HW, src, dst, DIS, ACC, E);
  combine_kernel<<<N, HIDDIM, 0, stream>>>(ACC, HW, DIS, bc2, H);

  // --- MLP head ---
  gather_kernel<<<NS, HIDDIM, 0, stream>>>(H, sgen, Z);
  // z1 = relu([Z, action] @ Wm1 + bm1): first 256 K-rows via WMMA, the 257th
  // (action column) folded into the epilogue as action[r] * Wm1[256,:].
  gemm_wmma_f32<<<gemmGridS, gemmBlk, 0, stream>>>(Z, Wm1, bm1, Z1, NS, HIDDIM,
                                                   HIDDIM, 1, action,
                                                   Wm1 + (size_t)HIDDIM * HIDDIM);
  // z2 = relu(z1 @ Wm2 + bm2)  (reuse Z buffer as Z2)
  gemm_wmma_f32<<<gemmGridS, gemmBlk, 0, stream>>>(Z1, Wm2, bm2, Z, NS, HIDDIM,
                                                   HIDDIM, 1, nullptr, nullptr);
  // q = mean(z2) @ Wh + bh
  reduce_q_kernel<<<1, HIDDIM, 0, stream>>>(Z, Wh, bh, (float*)d_out, NS);
}